// SBRLoss_48163763257928
// MI455X (gfx1250) — compile-verified
//
#include <hip/hip_runtime.h>
#include <hip/hip_bf16.h>
#include <math.h>

// Problem constants (from reference)
#define BATCH 4096
#define DIM   2048
#define NCLS  1000
#define CPAD  1024            // NCLS padded so WMMA tiles/wave is uniform (256 tiles = 32/wave)
#define CHUNK 16              // columns of z handled per workgroup
#define NCHUNK (DIM / CHUNK)  // 128 workgroups
#define STRIDE 17             // CHUNK+1: pad so WMMA column reads are bank-conflict-free
#define LDS_FLOATS (CPAD * STRIDE)        // 17408 floats
#define LDS_BYTES  (LDS_FLOATS * 4)       // 69632 B (< 320KB WGP LDS)

typedef __attribute__((ext_vector_type(2))) float v2f;
typedef __attribute__((ext_vector_type(8))) float v8f;

// ws layout: acc[0]=sup_sum, acc[1]=sum_i n_ti*||z_i||^2, acc[2]=sum_c ||m_c||^2
// cnt[0..CPAD) histogram at byte offset 64.

__global__ void sbr_init_kernel(float* acc, int* cnt) {
  int i = blockIdx.x * blockDim.x + threadIdx.x;
  if (i < 4) acc[i] = 0.0f;
  if (i < CPAD) cnt[i] = 0;
}

__global__ void sbr_hist_kernel(const int* __restrict__ targets, int* cnt) {
  int i = blockIdx.x * blockDim.x + threadIdx.x;
  if (i < BATCH) atomicAdd(&cnt[targets[i]], 1);
}

// One 256-thread block per row: cross-entropy via max + log-sum-exp.
// Row cached in 4 registers/thread so global traffic is one pass (16.4 MB).
__global__ void __launch_bounds__(256) sbr_ce_kernel(
    const float* __restrict__ outputs, const int* __restrict__ targets,
    float* acc) {
  __shared__ float red[256];
  const int row = blockIdx.x;
  const int tid = threadIdx.x;
  const float* x = outputs + (size_t)row * NCLS;

  float xv[4];
  float m = -3.402823466e38f;
#pragma unroll
  for (int j = 0; j < 4; ++j) {
    const int c = tid + j * 256;
    xv[j] = (c < NCLS) ? x[c] : -3.402823466e38f;
    m = fmaxf(m, xv[j]);
  }
  red[tid] = m;
  __syncthreads();
  for (int off = 128; off > 0; off >>= 1) {
    if (tid < off) red[tid] = fmaxf(red[tid], red[tid + off]);
    __syncthreads();
  }
  m = red[0];
  __syncthreads();

  float s = 0.0f;
#pragma unroll
  for (int j = 0; j < 4; ++j) {
    const int c = tid + j * 256;
    s += (c < NCLS) ? expf(xv[j] - m) : 0.0f;
  }
  red[tid] = s;
  __syncthreads();
  for (int off = 128; off > 0; off >>= 1) {
    if (tid < off) red[tid] += red[tid + off];
    __syncthreads();
  }
  if (tid == 0) {
    float lse = m + logf(red[0]);
    atomicAdd(&acc[0], lse - x[targets[row]]);   // -log p(target)
  }
}

// One workgroup per 16-column chunk of z. Streams z once (b128 loads); builds
// per-class column sums M_chunk[CPAD][CHUNK] in LDS with ds_add_f32 atomics,
// then uses V_WMMA_F32_16X16X4_F32 (D = A*A^T + C, diagonal = per-class sum of
// squares) to reduce sum_c ||m_c||^2 restricted to this chunk.
__global__ void __launch_bounds__(256) sbr_chunk_kernel(
    const float* __restrict__ z, const int* __restrict__ targets,
    const int* __restrict__ cnt, float* acc) {
  extern __shared__ float lds[];  // [CPAD][STRIDE], cols 0..CHUNK-1 used
  const int tid  = threadIdx.x;
  const int col0 = blockIdx.x * CHUNK;
  const int c4   = (tid & 3) << 2;           // 0,4,8,12: this thread's 4 columns
  const int rsub = tid >> 2;                 // 0..63: 64 rows per iteration

  for (int i = tid; i < LDS_FLOATS; i += 256) lds[i] = 0.0f;
  __syncthreads();

  float wsq = 0.0f;
  for (int row = rsub; row < BATCH; row += 64) {
    const float4 v =
        *(const float4*)&z[(size_t)row * DIM + col0 + c4];   // global_load_b128
    if (row + 64 < BATCH)
      __builtin_prefetch(&z[(size_t)(row + 64) * DIM + col0 + c4], 0, 0);
    const int t = targets[row];
    const float nf = (float)cnt[t];
    wsq += nf * (v.x * v.x + v.y * v.y + v.z * v.z + v.w * v.w);
    float* mrow = &lds[t * STRIDE + c4];
    atomicAdd(&mrow[0], v.x);                // ds_add_f32
    atomicAdd(&mrow[1], v.y);
    atomicAdd(&mrow[2], v.z);
    atomicAdd(&mrow[3], v.w);
  }
  __syncthreads();

  // WMMA phase: 256 tiles of 16 classes x 4 columns; exactly 32 tiles per
  // wave -> uniform counted loop, EXEC provably all-ones at every WMMA.
  // For the f32 16x16x4 op the 16x4 A operand layout and 4x16 B operand
  // layout are per-lane transposes, so the same register pair encodes A and
  // B = A^T.
  const int wave  = tid >> 5;                // 8 waves (wave32)
  const int lane  = tid & 31;
  const int mrow  = lane & 15;
  const int kbase = (lane >> 4) << 1;        // lanes 0-15: K=0,1; 16-31: K=2,3
  v8f cacc = {};
#pragma unroll 4
  for (int it = 0; it < 32; ++it) {
    const int tile = (it << 3) | wave;       // bijective over [0,256)
    const int r16  = (tile >> 2) << 4;       // class-tile base row
    const int k4   = (tile & 3) << 2;        // column-tile base
    const float* base = &lds[(r16 + mrow) * STRIDE + k4 + kbase];
    v2f a;
    a.x = base[0];
    a.y = base[1];
    cacc = __builtin_amdgcn_wmma_f32_16x16x4_f32(
        /*neg_a=*/false, a, /*neg_b=*/false, a,
        /*c_mod=*/(short)0, cacc, /*reuse_a=*/false, /*reuse_b=*/false);
  }
  // Diagonal of accumulated D: lanes 0..7 hold D[v][v] in cacc[v] (v==lane);
  // lanes 24..31 hold D[8+v][8+v] in cacc[v] (v==lane-24).
  float msq = 0.0f;
#pragma unroll
  for (int v = 0; v < 8; ++v)
    msq += ((lane == v) || (lane == 24 + v)) ? (float)cacc[v] : 0.0f;

  __syncthreads();                           // done reading M; reuse LDS front
  lds[tid] = wsq;
  lds[256 + tid] = msq;
  __syncthreads();
  for (int off = 128; off > 0; off >>= 1) {
    if (tid < off) {
      lds[tid] += lds[tid + off];
      lds[256 + tid] += lds[256 + tid + off];
    }
    __syncthreads();
  }
  if (tid == 0) {
    atomicAdd(&acc[1], lds[0]);
    atomicAdd(&acc[2], lds[256]);
  }
}

__global__ void sbr_finalize_kernel(const float* acc, float* out) {
  const float inv_b = 1.0f / (float)BATCH;
  const float sup = acc[0] * inv_b;
  const float reg = (acc[1] - acc[2]) * inv_b;
  out[0] = sup + 1e-5f * reg;
}

extern "C" void kernel_launch(void* const* d_in, const int* in_sizes, int n_in,
                              void* d_out, int out_size, void* d_ws, size_t ws_size,
                              hipStream_t stream) {
  const float* outputs = (const float*)d_in[0];   // [4096,1000] f32
  const float* z       = (const float*)d_in[1];   // [4096,2048] f32
  const int*   targets = (const int*)d_in[2];     // [4096] int
  float* out = (float*)d_out;

  float* acc = (float*)d_ws;
  int*   cnt = (int*)((char*)d_ws + 64);

  sbr_init_kernel<<<4, 256, 0, stream>>>(acc, cnt);
  sbr_hist_kernel<<<(BATCH + 255) / 256, 256, 0, stream>>>(targets, cnt);
  sbr_ce_kernel<<<BATCH, 256, 0, stream>>>(outputs, targets, acc);
  sbr_chunk_kernel<<<NCHUNK, 256, LDS_BYTES, stream>>>(z, targets, cnt, acc);
  sbr_finalize_kernel<<<1, 1, 0, stream>>>(acc, out);
}